// gat_adj_53601191854662
// MI455X (gfx1250) — compile-verified
//
#include <hip/hip_runtime.h>

typedef __attribute__((ext_vector_type(2))) float v2f;
typedef __attribute__((ext_vector_type(8))) float v8f;

#define N_    256
#define T_    64
#define V_    25
#define CIN_  3
#define K_    3
#define C_    64
#define ALPHA_ 0.2f
#define EPN_  (V_ * V_ * K_)   // 1875 elements of E / output per n

__global__ __launch_bounds__(128) void gat_adj_kernel(
    const float* __restrict__ x,      // (N, T, V, CIN)
    const float* __restrict__ Wconv,  // (CIN, C*K)
    const float* __restrict__ Wa,     // (2C, 1)
    float* __restrict__ out)          // (N, K, V, V)
{
  const int n    = blockIdx.x;
  const int tid  = threadIdx.x;
  const int lane = tid & 31;          // wave32
  const int wave = tid >> 5;

  __shared__ float xbar[V_ * CIN_ + 1];  // mean over t, 75 used
  __shared__ float Beff[CIN_][6];        // [i][col]; col 0..2 = Wa1eff, 3..5 = Wa2eff
  __shared__ float s1s[V_ * K_ + 1];
  __shared__ float s2s[V_ * K_ + 1];
  __shared__ float Ef[EPN_];
  __shared__ float rowsum[V_ * K_ + 1];

  // ---- Stage A: xbar[v*3+i] = mean_t x[n,t,v,i]  (coalesced: 32 consecutive e per wave)
  const float* xn = x + (size_t)n * (T_ * V_ * CIN_);
  for (int e = tid; e < V_ * CIN_; e += blockDim.x) {
    float s = 0.f;
    #pragma unroll 8
    for (int t = 0; t < T_; ++t) s += xn[t * (V_ * CIN_) + e];
    xbar[e] = s * (1.0f / T_);
  }
  // ---- Pre-contract W_conv with wa1/wa2: Beff[i][k] = sum_c Wconv[i, c*3+k]*Wa[c (or C+c)]
  if (tid < 18) {
    const int i = tid / 6, col = tid % 6, k = col % 3;
    const float* wa = Wa + (col < 3 ? 0 : C_);
    float s = 0.f;
    for (int c = 0; c < C_; ++c) s += Wconv[i * (C_ * K_) + c * K_ + k] * wa[c];
    Beff[i][col] = s;
  }
  __syncthreads();

  // ---- Stage B: s(25x6) = xbar(25x3, K padded to 4) * Beff(3x6, padded to 4x16)
  // via V_WMMA_F32_16X16X4_F32; tile 0 -> v=0..15 (wave 0), tile 1 -> v=16..24 (wave 1).
  if (wave < 2) {
    const int  base = wave * 16;
    const int  m    = lane & 15;
    const bool hi   = lane >= 16;   // lanes 16-31 carry K=2,3
    const int  v    = base + m;

    // A (16x4 f32): lane<16: VGPR0=K0, VGPR1=K1 ; lane>=16: VGPR0=K2, VGPR1=K3(pad 0)
    v2f a;
    a.x = (v < V_) ? xbar[v * 3 + (hi ? 2 : 0)] : 0.f;
    a.y = (v < V_ && !hi) ? xbar[v * 3 + 1] : 0.f;

    // B (4x16 f32): lane<16: VGPR0=B[K=0][N], VGPR1=B[K=1][N] ; lane>=16: VGPR0=B[K=2][N], VGPR1=B[K=3][N]=0
    v2f b;
    if (!hi) {
      b.x = (m < 6) ? Beff[0][m] : 0.f;
      b.y = (m < 6) ? Beff[1][m] : 0.f;
    } else {
      b.x = (m < 6) ? Beff[2][m] : 0.f;
      b.y = 0.f;
    }

    v8f c = {};
    v8f d = __builtin_amdgcn_wmma_f32_16x16x4_f32(
        /*neg_a=*/false, a, /*neg_b=*/false, b,
        /*c_mod=*/(short)0, c, /*reuse_a=*/false, /*reuse_b=*/false);

    // D layout: lane<16 -> (M=r, N=lane) ; lane>=16 -> (M=8+r, N=lane-16)
    #pragma unroll
    for (int r = 0; r < 8; ++r) {
      const int M  = r + (hi ? 8 : 0);
      const int vv = base + M;
      if (m < 6 && vv < V_) {
        const int k = m % 3;
        if (m < 3) s1s[vv * 3 + k] = d[r];
        else       s2s[vv * 3 + k] = d[r];
      }
    }
  }
  __syncthreads();

  // ---- Stage C: E flat (per-n shape (25,25,3), f = i*75 + j*3 + k) + leaky ReLU
  for (int f = tid; f < EPN_; f += blockDim.x) {
    const int i   = f / (V_ * K_);
    const int rem = f % (V_ * K_);
    const int j   = rem / K_;
    const int k   = rem % K_;
    const float e = s1s[j * 3 + k] + s2s[i * 3 + k];
    Ef[f] = (e >= 0.f) ? e : ALPHA_ * e;
  }
  __syncthreads();

  // ---- Stage D: softmax over each 25-contiguous-element row (75 rows), max-subtracted
  for (int row = tid; row < V_ * K_; row += blockDim.x) {
    const int o = row * V_;
    float mx = -3.402823466e38f;
    for (int b2 = 0; b2 < V_; ++b2) mx = fmaxf(mx, Ef[o + b2]);
    float s = 0.f;
    for (int b2 = 0; b2 < V_; ++b2) {
      const float ev = __expf(Ef[o + b2] - mx);
      Ef[o + b2] = ev;
      s += ev;
    }
    rowsum[row] = s;
  }
  __syncthreads();

  // ---- Stage E: normalized write-out, coalesced
  float* on = out + (size_t)n * EPN_;
  for (int f = tid; f < EPN_; f += blockDim.x) {
    on[f] = Ef[f] / rowsum[f / V_];
  }
}

extern "C" void kernel_launch(void* const* d_in, const int* in_sizes, int n_in,
                              void* d_out, int out_size, void* d_ws, size_t ws_size,
                              hipStream_t stream) {
  const float* x     = (const float*)d_in[0];
  const float* Wconv = (const float*)d_in[1];
  const float* Wa    = (const float*)d_in[2];
  float* out         = (float*)d_out;
  gat_adj_kernel<<<N_, 128, 0, stream>>>(x, Wconv, Wa, out);
}